// FeatureBuilder_82738249990279
// MI455X (gfx1250) — compile-verified
//
#include <hip/hip_runtime.h>
#include <hip/hip_bf16.h>
#include <math.h>

// ---------------------------------------------------------------------------
// FeatureBuilder for MI455X (gfx1250, wave32, WMMA).
//   feats = (g @ rbf_w + rbf_b + src_emb[an] + dst_emb[am]) / sqrt(3)
// g @ rbf_w is a [P,50]x[50,128] f32 GEMM -> V_WMMA_F32_16X16X4_F32, 13 steps.
// Epilogue staged through LDS; each (n,m,b) 512B row leaves as coalesced
// nontemporal global_store_b128. E templated so all strides constant-fold.
// ---------------------------------------------------------------------------

typedef __attribute__((ext_vector_type(2))) float v2f;
typedef __attribute__((ext_vector_type(4))) float v4f;
typedef __attribute__((ext_vector_type(8))) float v8f;

#define GG 50            // NUM_GAUSSIANS
#define KSTEPS 13        // ceil(50 / 4)
#define KP (4 * KSTEPS)  // 52, zero-padded K
#define MAX_E 128        // EMBED_DIM (module constant)
#define EPAD 4           // LDS row pad: keeps 16B alignment, shifts banks

// ---------------------------------------------------------------------------
// Kernel 1: build padded pos/anum (gather form, hazard-free) + padded_mask.
// ---------------------------------------------------------------------------
__global__ void pad_gather_kernel(const float* __restrict__ pos,
                                  const int* __restrict__ natoms,
                                  const int* __restrict__ anum,
                                  float* __restrict__ padded_pos,   // [B,N,3]
                                  int* __restrict__ padded_anum,    // [B,N]
                                  float* __restrict__ out_pmask,    // [B,N]
                                  int N, int B) {
  int idx = blockIdx.x * blockDim.x + threadIdx.x;
  if (idx >= B * N) return;
  int b = idx / N;
  int nn = idx - b * N;
  int off = 0;
  for (int i = 0; i < b; ++i) off += natoms[i];
  int nb = natoms[b];
  float x = 0.f, y = 0.f, z = 0.f;
  int a = 0;
  if (nn < nb) {
    int t = off + nn;
    x = pos[3 * t + 0];
    y = pos[3 * t + 1];
    z = pos[3 * t + 2];
    a = anum[t];
  }
  padded_pos[3 * (size_t)idx + 0] = x;
  padded_pos[3 * (size_t)idx + 1] = y;
  padded_pos[3 * (size_t)idx + 2] = z;
  padded_anum[idx] = a;
  out_pmask[idx] = (nn < nb) ? 1.0f : 0.0f;
}

// ---------------------------------------------------------------------------
// Kernel 2: feats via WMMA. Block = (E/16) waves; block handles (b, n, 16 m's).
// CE > 0: compile-time E (strides/offsets fold to immediates). CE == 0:
// runtime-E fallback.
// ---------------------------------------------------------------------------
template <int CE>
__global__ __launch_bounds__(256) void feats_wmma_kernel(
    const float* __restrict__ padded_pos,   // [B,N,3]
    const int* __restrict__ padded_anum,    // [B,N]
    const float* __restrict__ src_emb,      // [NE,E]
    const float* __restrict__ dst_emb,      // [NE,E]
    const float* __restrict__ rbf_w,        // [G,E]
    const float* __restrict__ rbf_b,        // [E]
    float* __restrict__ feats,              // [N,N,B,E]
    int N, int B, int Eparam, float delta, float coeff) {
  __shared__ float g_sh[16 * KP];  // 16 rows (m) x 52 padded gaussians
  __shared__ __align__(16) float out_sh[16 * (MAX_E + EPAD)];  // staged D

  const int E = (CE > 0) ? CE : Eparam;              // embed dim
  const int TPB = (CE > 0) ? (2 * CE) : blockDim.x;  // threads per block
  const int EP = E + EPAD;                           // LDS row pitch

  const int mtiles = N >> 4;
  int bt = blockIdx.x;
  int mt = bt % mtiles;
  int n = (bt / mtiles) % N;
  int b = bt / (mtiles * N);
  int m0 = mt << 4;

  // ---- Phase 1: cooperative exp(): g[row][k] into LDS (computed once,
  //      shared by all e-tile waves in the block).
  float pnx = padded_pos[3 * ((size_t)b * N + n) + 0];
  float pny = padded_pos[3 * ((size_t)b * N + n) + 1];
  float pnz = padded_pos[3 * ((size_t)b * N + n) + 2];
  for (int idx = threadIdx.x; idx < 16 * KP; idx += TPB) {
    int row = idx / KP;
    int k = idx - row * KP;
    const float* pm = &padded_pos[3 * ((size_t)b * N + m0 + row)];
    float dx = pm[0] - pnx, dy = pm[1] - pny, dz = pm[2] - pnz;
    float d = sqrtf(dx * dx + dy * dy + dz * dz);
    float g = 0.0f;
    if (k < GG) {
      float t = d - delta * (float)k;
      g = expf(coeff * t * t);
    }
    g_sh[idx] = g;
  }
  __syncthreads();

  // ---- Phase 2: per-wave WMMA over one 16-wide slice of E.
  int lane = threadIdx.x & 31;
  int wave = threadIdx.x >> 5;
  int lo = lane & 15;  // row (A) / col (B,D) within tile
  int hi = lane >> 4;  // K-pair selector (A,B) / row+8 selector (D)
  int e0 = wave << 4;
  int ecol = e0 + lo;

  // Preload all B fragments (rbf_w): per ISA layout, v0 holds K={4s,4s+2},
  // v1 holds K={4s+1,4s+3}, split across lane halves. Zero-pad K >= 50.
  v2f Bf[KSTEPS];
#pragma unroll
  for (int s = 0; s < KSTEPS; ++s) {
    int k0 = 4 * s + 2 * hi;
    Bf[s].x = (k0 < GG) ? rbf_w[(size_t)k0 * E + ecol] : 0.0f;
    Bf[s].y = (k0 + 1 < GG) ? rbf_w[(size_t)(k0 + 1) * E + ecol] : 0.0f;
  }

  v8f acc = {0.f, 0.f, 0.f, 0.f, 0.f, 0.f, 0.f, 0.f};
#pragma unroll
  for (int s = 0; s < KSTEPS; ++s) {
    // A fragment: lane(lo,hi) needs g[lo][4s+2hi], g[lo][4s+2hi+1]
    // (contiguous, 8B-aligned -> ds_load_b64 / ds_load_2addr_b64).
    v2f a = *(const v2f*)&g_sh[lo * KP + 4 * s + 2 * hi];
    acc = __builtin_amdgcn_wmma_f32_16x16x4_f32(
        /*neg_a=*/false, a, /*neg_b=*/false, Bf[s],
        /*c_mod=*/(short)0, acc, /*reuse_a=*/false, /*reuse_b=*/false);
  }

  // ---- Phase 3a: stage raw accumulators to LDS.
  // D layout: VGPR i, lanes0-15 -> row i, lanes16-31 -> row i+8; col = lo.
  {
    float* dst = &out_sh[(hi << 3) * EP + ecol];
#pragma unroll
    for (int i = 0; i < 8; ++i) {
      dst[i * EP] = acc[i];  // constant byte offsets when CE > 0
    }
  }
  __syncthreads();

  // ---- Phase 3b: fused epilogue + fully-coalesced 512B row stores.
  // Each (n,m,b) row of E floats is emitted with b128 nontemporal stores;
  // m is uniform per row, so emb/bias terms are clean vector loads.
  const float inv_s3 = 0.57735026918962576f;  // 1/sqrt(3)
  int sa = padded_anum[b * N + n];
  size_t base = (((size_t)n * N + m0) * B + b) * (size_t)E;  // scalar part
  const int q = E >> 2;                                      // float4s / row
  int rowstride = B * E;
#pragma unroll
  for (int idx = threadIdx.x; idx < 16 * q; idx += TPB) {
    int r = idx / q;
    int c = idx - r * q;
    int e = c << 2;
    int da = padded_anum[b * N + m0 + r];
    v4f av = *(const v4f*)&out_sh[r * EP + e];
    v4f dv = *(const v4f*)&dst_emb[(size_t)da * E + e];
    v4f sv = *(const v4f*)&src_emb[(size_t)sa * E + e];
    v4f bb = *(const v4f*)&rbf_b[e];
    v4f o = (av + bb + sv + dv) * inv_s3;
    __builtin_nontemporal_store(
        o, (v4f*)(feats + base + (size_t)(r * rowstride) + e));
  }
}

// ---------------------------------------------------------------------------
// Kernel 3: mask / dist / vec_hat (pure bandwidth; b fastest for coalescing).
// ---------------------------------------------------------------------------
__global__ void geom_kernel(const float* __restrict__ padded_pos,
                            const int* __restrict__ natoms,
                            float* __restrict__ out_mask,  // [N,N,B]
                            float* __restrict__ out_dist,  // [N,N,B]
                            float* __restrict__ out_vec,   // [N,N,B,3]
                            int N, int B) {
  size_t P = (size_t)N * N * B;
  size_t idx = (size_t)blockIdx.x * blockDim.x + threadIdx.x;
  if (idx >= P) return;
  int b = (int)(idx % B);
  size_t nm = idx / B;
  int m = (int)(nm % N);
  int n = (int)(nm / N);
  const float* pn = &padded_pos[3 * ((size_t)b * N + n)];
  const float* pm = &padded_pos[3 * ((size_t)b * N + m)];
  float dx = pm[0] - pn[0], dy = pm[1] - pn[1], dz = pm[2] - pn[2];
  float d = sqrtf(dx * dx + dy * dy + dz * dz);
  float inv = 1.0f / fmaxf(d, 1e-12f);
  int nb = natoms[b];
  __builtin_nontemporal_store((n < nb && m < nb) ? 1.0f : 0.0f, out_mask + idx);
  __builtin_nontemporal_store(d, out_dist + idx);
  __builtin_nontemporal_store(dx * inv, out_vec + 3 * idx + 0);
  __builtin_nontemporal_store(dy * inv, out_vec + 3 * idx + 1);
  __builtin_nontemporal_store(dz * inv, out_vec + 3 * idx + 2);
}

// ---------------------------------------------------------------------------
extern "C" void kernel_launch(void* const* d_in, const int* in_sizes, int n_in,
                              void* d_out, int out_size, void* d_ws,
                              size_t ws_size, hipStream_t stream) {
  const float* pos = (const float*)d_in[0];
  const int* natoms = (const int*)d_in[1];
  const int* anum = (const int*)d_in[2];
  const float* src_emb = (const float*)d_in[3];
  const float* dst_emb = (const float*)d_in[4];
  const float* rbf_w = (const float*)d_in[5];
  const float* rbf_b = (const float*)d_in[6];

  const int B = in_sizes[1];      // 8
  const int E = in_sizes[6];      // 128
  const int G = in_sizes[5] / E;  // 50
  // Solve N from out_size: N^2*B*(E+5) + B*N == out_size.
  int N = 256;
  for (long nn = 16; nn <= 4096; nn += 16) {
    long v = nn * nn * (long)B * (E + 5) + (long)B * nn;
    if (v == (long)out_size) { N = (int)nn; break; }
  }

  const float delta = 12.0f / (float)(G - 1);  // RBF_RADIUS / (G-1)
  const float coeff = -0.5f / (delta * delta);

  // Workspace layout.
  float* padded_pos = (float*)d_ws;                           // B*N*3 floats
  int* padded_anum = (int*)(padded_pos + (size_t)B * N * 3);  // B*N ints

  // Output layout (concatenated, return order).
  size_t P = (size_t)N * N * B;
  float* out = (float*)d_out;
  float* o_feats = out;                     // P*E
  float* o_mask = o_feats + P * (size_t)E;  // P
  float* o_pmask = o_mask + P;              // B*N
  float* o_dist = o_pmask + (size_t)B * N;  // P
  float* o_vec = o_dist + P;                // 3*P

  // 1) padded pos / anum / padded_mask
  {
    int threads = 256;
    int blocks = (B * N + threads - 1) / threads;
    pad_gather_kernel<<<blocks, threads, 0, stream>>>(
        pos, natoms, anum, padded_pos, padded_anum, o_pmask, N, B);
  }

  // 2) feats via WMMA: one block per (b, n, m-tile); one wave per 16 of E.
  {
    int waves = E / 16;             // 8 -> 256 threads
    int threads = waves * 32;
    int blocks = B * N * (N / 16);  // 32768 for the reference shapes
    if (E == 128) {
      feats_wmma_kernel<128><<<blocks, threads, 0, stream>>>(
          padded_pos, padded_anum, src_emb, dst_emb, rbf_w, rbf_b, o_feats, N,
          B, E, delta, coeff);
    } else {
      feats_wmma_kernel<0><<<blocks, threads, 0, stream>>>(
          padded_pos, padded_anum, src_emb, dst_emb, rbf_w, rbf_b, o_feats, N,
          B, E, delta, coeff);
    }
  }

  // 3) mask / dist / vec_hat
  {
    int threads = 256;
    long blocks = ((long)P + threads - 1) / threads;
    geom_kernel<<<(int)blocks, threads, 0, stream>>>(padded_pos, natoms, o_mask,
                                                     o_dist, o_vec, N, B);
  }
}